// OT_GNN_layer_18451179504148
// MI455X (gfx1250) — compile-verified
//
#include <hip/hip_runtime.h>
#include <hip/hip_bf16.h>

// ---------------- problem constants (match reference) ----------------
#define FIN    512
#define HID    64
#define KNEI   8
#define LSUB   9      // K_NEI + 1
#define NTPL   10
#define NTN    8
#define NCLS   6
#define ALPHA  0.5f
#define EPSR   0.1f
#define NOUTER 3
#define NSINK  5

typedef __attribute__((ext_vector_type(2))) float v2f;
typedef __attribute__((ext_vector_type(8))) float v8f;

// =====================================================================
// fp32 WMMA GEMM:  C[Nrows x Ncols] = A[Nrows x K] @ B[K x Ncols]
// block = 128 threads (4 waves). Each wave computes one 16x16 tile of a
// 16 x 64 output slab (Ncols == 64 always here). Nrows % 16 == 0.
// Uses V_WMMA_F32_16X16X4_F32, K-loop stepping by 4.
// =====================================================================
__global__ __launch_bounds__(128)
void gemm_f32_wmma(const float* __restrict__ A, const float* __restrict__ B,
                   float* __restrict__ C, int Nrows, int K, int Ncols) {
  const int lane = threadIdx.x & 31;
  const int wave = threadIdx.x >> 5;
  const int row0 = blockIdx.x * 16;
  const int col0 = wave * 16;
  const int lrow = lane & 15;   // 0..15
  const int hi   = lane >> 4;   // 0 or 1 (upper half-lanes hold K+2,K+3)

  v8f acc = {};
  for (int k = 0; k < K; k += 4) {
    const int ka = k + hi * 2;
    // A fragment: VGPR0 = A[M][ka], VGPR1 = A[M][ka+1]  (contiguous pair)
    const float* ap = A + (size_t)(row0 + lrow) * K + ka;
    v2f a = *(const v2f*)ap;
    // B fragment: VGPR0 = B[ka][N], VGPR1 = B[ka+1][N]
    const float* bp = B + (size_t)ka * Ncols + col0 + lrow;
    v2f b;
    b.x = bp[0];
    b.y = bp[Ncols];
    acc = __builtin_amdgcn_wmma_f32_16x16x4_f32(
        /*neg_a=*/false, a, /*neg_b=*/false, b,
        /*c_mod=*/(short)0, acc, /*reuse_a=*/false, /*reuse_b=*/false);
  }
  // C/D layout: VGPR v holds row (row0 + v + 8*hi), column col0+lrow
  #pragma unroll
  for (int v = 0; v < 8; ++v)
    C[(size_t)(row0 + hi * 8 + v) * Ncols + col0 + lrow] = acc[v];
}

// =====================================================================
// GCN normalization + propagation helpers
// =====================================================================
__global__ void deg_init_kernel(float* __restrict__ deg, int n) {
  int i = blockIdx.x * blockDim.x + threadIdx.x;
  if (i < n) deg[i] = 1.0f;  // self-loop
}

__global__ void deg_edge_kernel(const int* __restrict__ dst, float* __restrict__ deg, int e) {
  int i = blockIdx.x * blockDim.x + threadIdx.x;
  if (i < e) atomicAdd(&deg[dst[i]], 1.0f);
}

__global__ void rsqrt_kernel(const float* __restrict__ deg, float* __restrict__ dinv, int n) {
  int i = blockIdx.x * blockDim.x + threadIdx.x;
  if (i < n) dinv[i] = rsqrtf(deg[i]);
}

// out[i] = xw[i] * dinv[node]^2   (self-loop term; also serves as init)
__global__ void prop_init_kernel(const float* __restrict__ xw, const float* __restrict__ dinv,
                                 float* __restrict__ out, int n) {
  int i = blockIdx.x * blockDim.x + threadIdx.x;
  if (i < n * HID) {
    int node = i / HID;
    float d = dinv[node];
    out[i] = xw[i] * d * d;
  }
}

// out[dst] += xw[src] * dinv[src]*dinv[dst], one thread per (edge, feature)
__global__ void prop_edge_kernel(const int* __restrict__ src, const int* __restrict__ dst,
                                 const float* __restrict__ dinv, const float* __restrict__ xw,
                                 float* __restrict__ out, int e) {
  int i = blockIdx.x * blockDim.x + threadIdx.x;
  if (i < e * HID) {
    int ed = i / HID, f = i % HID;
    int s = src[ed], d = dst[ed];
    float c = dinv[s] * dinv[d];
    atomicAdd(&out[(size_t)d * HID + f], xw[(size_t)s * HID + f] * c);
  }
}

__global__ void bias_act_kernel(float* __restrict__ h, const float* __restrict__ b,
                                int n, int do_relu) {
  int i = blockIdx.x * blockDim.x + threadIdx.x;
  if (i < n * HID) {
    float v = h[i] + b[i % HID];
    h[i] = do_relu ? fmaxf(v, 0.0f) : v;
  }
}

// =====================================================================
// Per-node fused-GW distance to templates. One 96-thread block per node.
// Tiny 9x8 problems live entirely in LDS; thread tid<72 owns entry (l,m).
// =====================================================================
__global__ __launch_bounds__(96)
void dist_tpl_kernel(const float* __restrict__ h,     // [N, HID]
                     const int*   __restrict__ nbr,   // [N, KNEI]
                     const float* __restrict__ mask,  // [N, LSUB]
                     const float* __restrict__ Cl,    // [N, LSUB, LSUB]
                     const float* __restrict__ tpl,   // [NTPL, NTN, NTN]
                     const float* __restrict__ tplf,  // [NTPL, NTN, HID]
                     float* __restrict__ y) {         // [N, NTPL]
  __shared__ float s_xl[LSUB][HID];
  __shared__ float s_xsq[LSUB];
  __shared__ float s_C1[LSUB][LSUB];
  __shared__ float s_h1[LSUB];
  __shared__ float s_rc1[LSUB];
  __shared__ float s_C2[NTN][NTN];
  __shared__ float s_rc2[NTN];
  __shared__ float s_tfsq[NTN];
  __shared__ float s_M[LSUB][NTN];
  __shared__ float s_Km[LSUB][NTN];
  __shared__ float s_T[LSUB][NTN];
  __shared__ float s_A[LSUB][NTN];
  __shared__ float s_u[LSUB];
  __shared__ float s_v[NTN];
  __shared__ float s_red[LSUB * NTN];
  __shared__ int   s_idx[LSUB];
  __shared__ float s_msum;

  const int n = blockIdx.x;
  const int tid = threadIdx.x;
  const float h2w = 1.0f / (float)NTN;

  if (tid < LSUB) s_idx[tid] = (tid == 0) ? n : nbr[(size_t)n * KNEI + tid - 1];
  if (tid == 0) {
    float s = 0.f;
    for (int l = 0; l < LSUB; ++l) s += mask[(size_t)n * LSUB + l];
    s_msum = s;
  }
  __syncthreads();

  for (int i = tid; i < LSUB * HID; i += 96) {
    int l = i / HID, f = i % HID;
    s_xl[l][f] = h[(size_t)s_idx[l] * HID + f];
  }
  for (int i = tid; i < LSUB * LSUB; i += 96)
    s_C1[i / LSUB][i % LSUB] = Cl[(size_t)n * LSUB * LSUB + i];
  if (tid < LSUB) s_h1[tid] = mask[(size_t)n * LSUB + tid] / s_msum;
  __syncthreads();

  if (tid < LSUB) {
    float s = 0.f;
    for (int f = 0; f < HID; ++f) { float v = s_xl[tid][f]; s += v * v; }
    s_xsq[tid] = s;
    float r = 0.f;
    for (int j = 0; j < LSUB; ++j) { float c = s_C1[tid][j]; r += c * c * s_h1[j]; }
    s_rc1[tid] = r;
  }
  __syncthreads();

  for (int t = 0; t < NTPL; ++t) {
    if (tid < NTN * NTN) s_C2[tid / NTN][tid % NTN] = tpl[(size_t)t * NTN * NTN + tid];
    __syncthreads();
    if (tid < NTN) {
      float r = 0.f, q = 0.f;
      for (int j = 0; j < NTN; ++j) { float c = s_C2[tid][j]; r += c * c; }
      s_rc2[tid] = r * h2w;
      const float* tf = tplf + ((size_t)t * NTN + tid) * HID;
      for (int f = 0; f < HID; ++f) { float v = tf[f]; q += v * v; }
      s_tfsq[tid] = q;
    }
    __syncthreads();
    if (tid < LSUB * NTN) {
      int l = tid / NTN, m = tid % NTN;
      const float* tf = tplf + ((size_t)t * NTN + m) * HID;
      float dot = 0.f;
      for (int f = 0; f < HID; ++f) dot += s_xl[l][f] * tf[f];
      s_M[l][m] = s_xsq[l] + s_tfsq[m] - 2.0f * dot;
      s_T[l][m] = s_h1[l] * h2w;
    }
    __syncthreads();

    for (int outer = 0; outer < NOUTER; ++outer) {
      if (tid < LSUB * NTN) {          // A = C1 @ T
        int l = tid / NTN, m = tid % NTN;
        float s = 0.f;
        for (int j = 0; j < LSUB; ++j) s += s_C1[l][j] * s_T[j][m];
        s_A[l][m] = s;
      }
      __syncthreads();
      if (tid < LSUB * NTN) {          // K = exp(-G/eps), G = (1-a)M + a*tens
        int l = tid / NTN, m = tid % NTN;
        float s = 0.f;
        for (int j = 0; j < NTN; ++j) s += s_A[l][j] * s_C2[j][m];
        float tens = s_rc1[l] + s_rc2[m] - 2.0f * s;
        float G = (1.0f - ALPHA) * s_M[l][m] + ALPHA * tens;
        s_Km[l][m] = __expf(-G / EPSR);
      }
      if (tid < NTN) s_v[tid] = 1.0f;
      __syncthreads();
      for (int it = 0; it < NSINK; ++it) {
        if (tid < LSUB) {
          float s = 0.f;
          for (int m = 0; m < NTN; ++m) s += s_Km[tid][m] * s_v[m];
          s_u[tid] = s_h1[tid] / (s + 1e-16f);
        }
        __syncthreads();
        if (tid < NTN) {
          float s = 0.f;
          for (int l = 0; l < LSUB; ++l) s += s_Km[l][tid] * s_u[l];
          s_v[tid] = h2w / (s + 1e-16f);
        }
        __syncthreads();
      }
      if (tid < LSUB * NTN) {
        int l = tid / NTN, m = tid % NTN;
        s_T[l][m] = s_u[l] * s_Km[l][m] * s_v[m];
      }
      __syncthreads();
    }

    // final objective: sum(T * ((1-a)M + a*(constC - 2 C1 T C2)))
    if (tid < LSUB * NTN) {
      int l = tid / NTN, m = tid % NTN;
      float s = 0.f;
      for (int j = 0; j < LSUB; ++j) s += s_C1[l][j] * s_T[j][m];
      s_A[l][m] = s;
    }
    __syncthreads();
    if (tid < LSUB * NTN) {
      int l = tid / NTN, m = tid % NTN;
      float s = 0.f;
      for (int j = 0; j < NTN; ++j) s += s_A[l][j] * s_C2[j][m];
      float tens = s_rc1[l] + s_rc2[m] - 2.0f * s;
      s_red[tid] = s_T[l][m] * ((1.0f - ALPHA) * s_M[l][m] + ALPHA * tens);
    }
    __syncthreads();
    if (tid == 0) {
      float s = 0.f;
      for (int i = 0; i < LSUB * NTN; ++i) s += s_red[i];
      y[(size_t)n * NTPL + t] = s;
    }
    __syncthreads();
  }
}

// =====================================================================
// Final: out = relu(concat(h, y) @ Wlin + blin)
// =====================================================================
__global__ void final_linear_kernel(const float* __restrict__ h, const float* __restrict__ y,
                                    const float* __restrict__ Wlin, const float* __restrict__ blin,
                                    float* __restrict__ out, int n) {
  int i = blockIdx.x * blockDim.x + threadIdx.x;
  if (i >= n * NCLS) return;
  int node = i / NCLS, c = i % NCLS;
  float s = blin[c];
  const float* hp = h + (size_t)node * HID;
  for (int k = 0; k < HID; ++k) s += hp[k] * Wlin[(size_t)k * NCLS + c];
  const float* yp = y + (size_t)node * NTPL;
  for (int t = 0; t < NTPL; ++t) s += yp[t] * Wlin[(size_t)(HID + t) * NCLS + c];
  out[i] = fmaxf(s, 0.0f);
}

// =====================================================================
extern "C" void kernel_launch(void* const* d_in, const int* in_sizes, int n_in,
                              void* d_out, int out_size, void* d_ws, size_t ws_size,
                              hipStream_t stream) {
  const float* x    = (const float*)d_in[0];   // [N, FIN]
  const int*   ei   = (const int*)d_in[1];     // [2, E]
  const int*   nbr  = (const int*)d_in[2];     // [N, KNEI]
  const float* mask = (const float*)d_in[3];   // [N, LSUB]
  const float* Cl   = (const float*)d_in[4];   // [N, LSUB, LSUB]
  const float* W1   = (const float*)d_in[5];
  const float* b1   = (const float*)d_in[6];
  const float* W2   = (const float*)d_in[7];
  const float* b2   = (const float*)d_in[8];
  const float* tpl  = (const float*)d_in[9];
  const float* tplf = (const float*)d_in[10];
  const float* Wlin = (const float*)d_in[11];
  const float* blin = (const float*)d_in[12];
  float* out = (float*)d_out;

  const int N = in_sizes[2] / KNEI;   // 20000
  const int E = in_sizes[1] / 2;      // 320000
  const int* src = ei;
  const int* dst = ei + E;

  // workspace carve-up (~11.3 MB)
  char* w = (char*)d_ws;
  float* deg  = (float*)w;  w += (size_t)N * sizeof(float);
  float* dinv = (float*)w;  w += (size_t)N * sizeof(float);
  float* bufA = (float*)w;  w += (size_t)N * HID * sizeof(float);
  float* bufB = (float*)w;  w += (size_t)N * HID * sizeof(float);
  float* yb   = (float*)w;  w += (size_t)N * NTPL * sizeof(float);

  const int TB = 256;
  // degree + D^-1/2 (shared by both GCN layers)
  deg_init_kernel<<<(N + TB - 1) / TB, TB, 0, stream>>>(deg, N);
  deg_edge_kernel<<<(E + TB - 1) / TB, TB, 0, stream>>>(dst, deg, E);
  rsqrt_kernel<<<(N + TB - 1) / TB, TB, 0, stream>>>(deg, dinv, N);

  // layer 1: xw = x @ W1 (WMMA fp32), propagate, +b1, ReLU
  gemm_f32_wmma<<<N / 16, 128, 0, stream>>>(x, W1, bufA, N, FIN, HID);
  prop_init_kernel<<<((size_t)N * HID + TB - 1) / TB, TB, 0, stream>>>(bufA, dinv, bufB, N);
  prop_edge_kernel<<<((size_t)E * HID + TB - 1) / TB, TB, 0, stream>>>(src, dst, dinv, bufA, bufB, E);
  bias_act_kernel<<<((size_t)N * HID + TB - 1) / TB, TB, 0, stream>>>(bufB, b1, N, 1);

  // layer 2: xw = h1 @ W2 (WMMA fp32), propagate, +b2
  gemm_f32_wmma<<<N / 16, 128, 0, stream>>>(bufB, W2, bufA, N, HID, HID);
  prop_init_kernel<<<((size_t)N * HID + TB - 1) / TB, TB, 0, stream>>>(bufA, dinv, bufB, N);
  prop_edge_kernel<<<((size_t)E * HID + TB - 1) / TB, TB, 0, stream>>>(src, dst, dinv, bufA, bufB, E);
  bias_act_kernel<<<((size_t)N * HID + TB - 1) / TB, TB, 0, stream>>>(bufB, b2, N, 0);

  // fused-GW distance to templates (one block per node)
  dist_tpl_kernel<<<N, 96, 0, stream>>>(bufB, nbr, mask, Cl, tpl, tplf, yb);

  // final linear + relu
  final_linear_kernel<<<((size_t)N * NCLS + TB - 1) / TB, TB, 0, stream>>>(bufB, yb, Wlin, blin, out, N);
}